// SpeedPerturb_13872744366255
// MI455X (gfx1250) — compile-verified
//
#include <hip/hip_runtime.h>
#include <stdint.h>

typedef __attribute__((ext_vector_type(2))) float v2f;
typedef __attribute__((ext_vector_type(8))) float v8f;

#define N_IN      480000
#define N_OUT     432000
#define PHASES    9
#define STRIDE    10
#define TAPS      14
#define NBATCH    16
#define TILE_OUT  144            // outputs per macro-tile per batch (= lcm(16,9))
#define TILE_IN   160            // input advance per macro-tile
#define NTILES    (N_OUT / TILE_OUT)        // 3000
#define WAVES_PER_BLOCK 8
#define BLOCK     (WAVES_PER_BLOCK * 32)
#define GRID      (NTILES / WAVES_PER_BLOCK) // 375, exact
#define WAVROW    196            // LDS row stride (floats) for staged waveform
#define WAVFLOATS (NBATCH * WAVROW)
#define OUTROW    148            // LDS row stride (floats) for output transpose
#define AFRAGS    (9 * 8 * 32)   // 9 sub-tiles * 8 k-steps * 32 lanes

__device__ __forceinline__ int startOf(int n) {
  const int fi[PHASES] = {-6, -5, -4, -3, -2, -1, 0, 2, 3};
  return fi[n % PHASES] + STRIDE * (n / PHASES);
}

// low 32 bits of a generic pointer into LDS == wave-relative LDS byte offset
__device__ __forceinline__ unsigned ldsOff(const void* p) {
  return (unsigned)(uintptr_t)p;
}

__global__ void resample_wmma_f32(const float* __restrict__ wav,
                                  const float* __restrict__ wgt,
                                  float* __restrict__ out) {
  extern __shared__ char smem[];
  v2f*   sA      = (v2f*)smem;                              // 2304 float2 = 18.4 KB
  float* sWavAll = (float*)(smem + AFRAGS * sizeof(v2f));   // 8 waves * 16x196 f32

  const int tid  = threadIdx.x;
  const int lane = tid & 31;
  const int wv   = tid >> 5;
  const int tile = blockIdx.x * WAVES_PER_BLOCK + wv;       // 0..2999, no tail
  float* sW = sWavAll + wv * WAVFLOATS;

  // ---- Precompute pre-swizzled A fragments (identical for every tile) ----
  for (int e = tid; e < AFRAGS; e += BLOCK) {
    int l    = e & 31;
    int kb   = (e >> 5) & 7;
    int j    = e >> 8;                 // sub-tile 0..8
    int u    = l & 15;                 // output row within sub-tile
    int half = l >> 4;
    int n0   = 16 * j;
    int s0   = startOf(n0);
    int n    = n0 + u;
    int p    = n % PHASES;
    int d    = startOf(n) - s0;        // 0..17, so d+13 <= 31 fits K=32
    int ks   = 4 * kb + 2 * half;      // A layout: lanes=M, VGPR0/1 hold K {0,2}/{1,3}
    float ax = 0.f, ay = 0.f;
    int w0 = ks - d;
    if (w0 >= 0 && w0 < TAPS) ax = wgt[p * TAPS + w0];
    int w1 = ks + 1 - d;
    if (w1 >= 0 && w1 < TAPS) ay = wgt[p * TAPS + w1];
    v2f a; a.x = ax; a.y = ay;
    sA[e] = a;
  }

  // ---- Stage this wave's input window [160*tile-16, 160*tile+176) into LDS ----
  const int G0 = tile * TILE_IN - 16;  // 16B-aligned global start
  if (G0 >= 0 && G0 + 192 <= N_IN) {
    // fast path: async global->LDS, 512B per instruction, ASYNCcnt-tracked
#pragma unroll
    for (int it = 0; it < 24; ++it) {  // 16 rows * 48 float4
      int f4  = it * 32 + lane;
      int row = f4 / 48;
      int c4  = f4 % 48;
      unsigned goff = (unsigned)((row * N_IN + G0 + c4 * 4) * 4);   // byte voffset
      unsigned loff = ldsOff(sW + row * WAVROW + c4 * 4);           // LDS byte addr
      asm volatile("global_load_async_to_lds_b128 %0, %1, %2"
                   :: "v"(loff), "v"(goff), "s"(wav) : "memory");
    }
    asm volatile("s_wait_asynccnt 0" ::: "memory");
  } else {                             // boundary tiles: element-wise with zero fill
    for (int e = lane; e < NBATCH * 192; e += 32) {
      int row = e / 192, col = e % 192;
      int g = G0 + col;
      float v = (g >= 0 && g < N_IN) ? wav[(size_t)row * N_IN + g] : 0.f;
      sW[row * WAVROW + col] = v;
    }
  }

  __syncthreads();

  // ---- 9 sub-tiles: D[16 outputs x 16 batches] += A[16x32] * B[32x16] ----
  const int c    = lane & 15;          // batch = N column
  const int half = lane >> 4;
  const int baseRel[PHASES] = {10, 28, 45, 63, 81, 99, 116, 134, 152}; // startOf(16j)+16

  v8f acc[9];
#pragma unroll
  for (int j = 0; j < 9; ++j) {
    v8f d = {};
    const float* wrow = sW + c * WAVROW + baseRel[j] + 2 * half;
#pragma unroll
    for (int kb = 0; kb < 8; ++kb) {
      v2f a = sA[(j * 8 + kb) * 32 + lane];
      v2f b; b.x = wrow[4 * kb]; b.y = wrow[4 * kb + 1];  // B: lanes=N, K {0,2}/{1,3}
      d = __builtin_amdgcn_wmma_f32_16x16x4_f32(false, a, false, b,
                                                (short)0, d, false, false);
    }
    acc[j] = d;
  }

  __syncthreads();  // input staging no longer needed; safe to reuse sW

  // ---- Transpose D tiles through LDS: rows=batch, 144 contiguous outputs ----
#pragma unroll
  for (int j = 0; j < 9; ++j) {
#pragma unroll
    for (int v = 0; v < 8; ++v) {
      int u = 16 * j + v + 8 * half;   // D layout: VGPR v -> M = v + 8*half
      sW[c * OUTROW + u] = acc[j][v];
    }
  }

  __syncthreads();
  asm volatile("s_wait_dscnt 0" ::: "memory");  // transpose writes visible to async reads

  // ---- Coalesced async LDS->global stores, 512B per instruction ----
#pragma unroll
  for (int it = 0; it < 18; ++it) {    // 16 rows * 36 float4
    int f4  = it * 32 + lane;
    int row = f4 / 36;
    int c4  = f4 % 36;
    unsigned goff = (unsigned)((row * N_OUT + tile * TILE_OUT + c4 * 4) * 4);
    unsigned loff = ldsOff(sW + row * OUTROW + c4 * 4);
    asm volatile("global_store_async_from_lds_b128 %0, %1, %2"
                 :: "v"(goff), "v"(loff), "s"(out) : "memory");
  }
  // s_endpgm performs an implicit wait-idle, covering outstanding ASYNCcnt
}

extern "C" void kernel_launch(void* const* d_in, const int* in_sizes, int n_in,
                              void* d_out, int out_size, void* d_ws, size_t ws_size,
                              hipStream_t stream) {
  const float* wav = (const float*)d_in[0];   // (16, 480000) f32
  const float* wgt = (const float*)d_in[1];   // (9, 14) f32
  float* out = (float*)d_out;                 // (16, 432000) f32
  const size_t smem = AFRAGS * sizeof(v2f) +
                      (size_t)WAVES_PER_BLOCK * WAVFLOATS * sizeof(float); // 118784 B
  resample_wmma_f32<<<GRID, BLOCK, smem, stream>>>(wav, wgt, out);
}